// DeepseekV4Compressor_78752520339581
// MI455X (gfx1250) — compile-verified
//
#include <hip/hip_runtime.h>
#include <hip/hip_bf16.h>

// ---------------- problem constants (from reference) ----------------
#define TT   8192            // tokens
#define HH   4096            // hidden
#define DD   512             // D
#define TWOD 1024            // 2*D
#define HD   256
#define QB   128             // quant block
#define BS64 64              // BS
#define LBT  32              // L/BS = 2048/64 block-table cols

// ---------------- GEMM tiling ----------------
#define WG_M 128
#define WG_N 128
#define KC   32
#define NCHUNK (HH / KC)     // 128
#define LDSW 36              // LDS row stride in floats (144B rows: 16B aligned, conflict-free frags)

typedef float v2f __attribute__((ext_vector_type(2)));
typedef float v8f __attribute__((ext_vector_type(8)));
typedef int   v4i_g __attribute__((vector_size(16)));   // GCC-style, matches builtin proto

#define AS_GLOBAL __attribute__((address_space(1)))
#define AS_LOCAL  __attribute__((address_space(3)))

#if __has_builtin(__builtin_amdgcn_global_load_async_to_lds_b128)
#define HAVE_ASYNC 1
#else
#define HAVE_ASYNC 0
#endif

__device__ __forceinline__ void wait_async_zero() {
#if HAVE_ASYNC
#if __has_builtin(__builtin_amdgcn_s_wait_asynccnt)
    __builtin_amdgcn_s_wait_asynccnt(0);
#else
    asm volatile("s_wait_asynccnt 0x0" ::: "memory");
#endif
#endif
}

// copy one 128x32 tile of A and of W into LDS (unscaled)
__device__ __forceinline__ void stage_tile(const float* __restrict__ gA,
                                           const float* __restrict__ gW,
                                           float* As, float* Bs,
                                           int lrow, int lcol, int k0)
{
    #pragma unroll
    for (int r = 0; r < 4; ++r) {
        const int row = lrow + r * 32;
#if HAVE_ASYNC
        __builtin_amdgcn_global_load_async_to_lds_b128(
            (AS_GLOBAL v4i_g*)(unsigned long long)(gA + (size_t)row * HH + k0 + lcol),
            (AS_LOCAL  v4i_g*)(unsigned long long)(As + row * LDSW + lcol), 0, 0);
        __builtin_amdgcn_global_load_async_to_lds_b128(
            (AS_GLOBAL v4i_g*)(unsigned long long)(gW + (size_t)row * HH + k0 + lcol),
            (AS_LOCAL  v4i_g*)(unsigned long long)(Bs + row * LDSW + lcol), 0, 0);
#else
        const float4 va = *(const float4*)(gA + (size_t)row * HH + k0 + lcol);
        *(float4*)(As + row * LDSW + lcol) = va;
        const float4 vb = *(const float4*)(gW + (size_t)row * HH + k0 + lcol);
        *(float4*)(Bs + row * LDSW + lcol) = vb;
#endif
    }
}

// kv_score = hidden @ (weight*scale)^T, fused epilogue:
//   kv -> d_kv, score -> d_score, state=[kv+ape[pos%4], score] scattered to d_sc[out_loc]
__global__ __launch_bounds__(256)
void gemm_fused_kernel(const float* __restrict__ A,        // hidden [8192][4096]
                       const float* __restrict__ W,        // weight [1024][4096]
                       const float* __restrict__ Wscale,   // [8][32]
                       const float* __restrict__ ape,      // [4][512]
                       const int*   __restrict__ positions,// [8192]
                       const int*   __restrict__ out_loc,  // [8192]
                       float* __restrict__ d_kv,           // [8192][512]
                       float* __restrict__ d_score,        // [8192][512]
                       float* __restrict__ d_sc)           // [8192][1024]
{
    __shared__ float AsB[2][WG_M * LDSW];
    __shared__ float BsB[2][WG_N * LDSW];

    const int tid  = threadIdx.x;
    const int lane = tid & 31;
    const int wv   = tid >> 5;          // 0..7
    const int wm   = wv & 3;            // 4 waves along M (32 rows each)
    const int wn   = wv >> 2;           // 2 waves along N (64 cols each)

    const int m_base = blockIdx.x * WG_M;
    const int n_base = blockIdx.y * WG_N;

    const int half = lane >> 4;         // lane groups of 16
    const int l16  = lane & 15;
    const int koff = half * 2;          // A/B frag: lanes 16-31 hold K+2,K+3

    const float* gA = A + (size_t)m_base * HH;
    const float* gW = W + (size_t)n_base * HH;

    const v8f zerov = (v8f){0.f,0.f,0.f,0.f,0.f,0.f,0.f,0.f};
    v8f acc[2][4];   // master (scaled) accumulators
    v8f part[2][4];  // per-128K-block unscaled partials
    #pragma unroll
    for (int i = 0; i < 2; ++i)
        #pragma unroll
        for (int j = 0; j < 4; ++j) { acc[i][j] = zerov; part[i][j] = zerov; }

    // cooperative staging map: 256 threads x (4 rows A + 4 rows B) x float4
    const int lrow = tid >> 3;          // 0..31
    const int lcol = (tid & 7) * 4;     // 0..28

    const int scale_row = (n_base >> 7) * (HH / QB);   // weight_scale row base

    stage_tile(gA, gW, AsB[0], BsB[0], lrow, lcol, 0);

    for (int c = 0; c < NCHUNK; ++c) {
        wait_async_zero();              // my copies for buf[c&1] are in LDS
        __syncthreads();                // everyone's copies done; prev reads of next buf done
        if (c + 1 < NCHUNK)
            stage_tile(gA, gW, AsB[(c + 1) & 1], BsB[(c + 1) & 1], lrow, lcol, (c + 1) * KC);

        const float* cAs = AsB[c & 1];
        const float* cBs = BsB[c & 1];

        #pragma unroll
        for (int kk = 0; kk < KC; kk += 4) {
            v2f afrag[2], bfrag[4];
            #pragma unroll
            for (int mi = 0; mi < 2; ++mi)
                afrag[mi] = *(const v2f*)(cAs + (wm*32 + mi*16 + l16) * LDSW + kk + koff);
            #pragma unroll
            for (int ni = 0; ni < 4; ++ni)
                bfrag[ni] = *(const v2f*)(cBs + (wn*64 + ni*16 + l16) * LDSW + kk + koff);
            #pragma unroll
            for (int mi = 0; mi < 2; ++mi)
                #pragma unroll
                for (int ni = 0; ni < 4; ++ni)
                    part[mi][ni] = __builtin_amdgcn_wmma_f32_16x16x4_f32(
                        false, afrag[mi], false, bfrag[ni],
                        (short)0, part[mi][ni], false, false);
        }

        if ((c & 3) == 3) {             // end of a 128-wide K quant block: apply scale
            const float s = Wscale[scale_row + (c >> 2)];
            #pragma unroll
            for (int mi = 0; mi < 2; ++mi)
                #pragma unroll
                for (int ni = 0; ni < 4; ++ni) {
                    acc[mi][ni] = acc[mi][ni] + part[mi][ni] * s;
                    part[mi][ni] = zerov;
                }
        }
    }

    // epilogue: C/D layout -> VGPR r: row = r + 8*half, col = l16
    #pragma unroll
    for (int mi = 0; mi < 2; ++mi) {
        #pragma unroll
        for (int ni = 0; ni < 4; ++ni) {
            #pragma unroll
            for (int r = 0; r < 8; ++r) {
                const int gm = m_base + wm*32 + mi*16 + half*8 + r;
                const int gn = n_base + wn*64 + ni*16 + l16;
                const float v = acc[mi][ni][r];
                const int loc = out_loc[gm];
                float stv;
                if (gn < DD) {
                    d_kv[(size_t)gm * DD + gn] = v;
                    stv = v + ape[(positions[gm] & 3) * DD + gn];
                } else {
                    d_score[(size_t)gm * DD + (gn - DD)] = v;
                    stv = v;
                }
                d_sc[(size_t)loc * TWOD + gn] = stv;
            }
        }
    }
}

// masked tokens only: window gather -> per-channel softmax over window -> RMS -> RoPE -> kc
__global__ __launch_bounds__(256)
void compose_kernel(const float* __restrict__ sc,          // [8192][1024]
                    const float* __restrict__ norm_w,      // [256]
                    const float* __restrict__ cs_cache,    // [512][256]
                    const int*   __restrict__ positions,   // [8192]
                    const int*   __restrict__ t2r,         // [8192]
                    const int*   __restrict__ btab,        // [4][32]
                    const int*   __restrict__ kvmap,       // [8192]
                    float* __restrict__ kc)                // [2048][512]
{
    const int t   = blockIdx.x;
    const int pos = positions[t];
    if ((pos & 3) != 3) return;          // only window-closing tokens write kv_cache
    const int tid = threadIdx.x;
    const int req = t2r[t];

    __shared__ float xa[512];
    __shared__ float red[256];
    __shared__ int   slots_s[4];

    if (tid < 4) {
        const int wp = pos - 3 + tid;    // pos%4==3 -> wp >= 0 always
        slots_s[tid] = btab[req * LBT + (wp >> 6)] * BS64 + (wp & 63);
    }
    __syncthreads();
    const int s0 = slots_s[0], s1 = slots_s[1], s2 = slots_s[2], s3 = slots_s[3];

    #pragma unroll
    for (int h = 0; h < 2; ++h) {
        const int c = tid + h * 256;
        // softmax over the 4-row window, per channel
        const float g0 = sc[(size_t)s0 * TWOD + DD + c];
        const float g1 = sc[(size_t)s1 * TWOD + DD + c];
        const float g2 = sc[(size_t)s2 * TWOD + DD + c];
        const float g3 = sc[(size_t)s3 * TWOD + DD + c];
        const float mx = fmaxf(fmaxf(g0, g1), fmaxf(g2, g3));
        const float e0 = expf(g0 - mx), e1 = expf(g1 - mx);
        const float e2 = expf(g2 - mx), e3 = expf(g3 - mx);
        const float den = e0 + e1 + e2 + e3;
        const float k0v = sc[(size_t)s0 * TWOD + c];
        const float k1v = sc[(size_t)s1 * TWOD + c];
        const float k2v = sc[(size_t)s2 * TWOD + c];
        const float k3v = sc[(size_t)s3 * TWOD + c];
        const float comp = (e0*k0v + e1*k1v + e2*k2v + e3*k3v) / den;

        // RMS over this 256-channel half
        red[tid] = comp * comp;
        __syncthreads();
        #pragma unroll
        for (int off = 128; off > 0; off >>= 1) {
            if (tid < off) red[tid] += red[tid + off];
            __syncthreads();
        }
        const float inv = rsqrtf(red[0] * (1.0f / 256.0f) + 1e-6f);
        __syncthreads();                  // allow red[] reuse next half
        xa[c] = comp * inv * norm_w[tid];
    }
    __syncthreads();

    // RoPE (half-split rotation) + scatter to kv_cache slot
    const float* cs = cs_cache + (size_t)(pos >> 2) * HD;  // [cos(128) | sin(128)]
    const int slot = kvmap[t];
    #pragma unroll
    for (int h = 0; h < 2; ++h) {
        const int c = tid + h * 256;
        const int i = tid;               // index within HD half-block
        float o;
        if (i < 128) {
            o = xa[h*256 + i] * cs[i] - xa[h*256 + i + 128] * cs[128 + i];
        } else {
            const int ii = i - 128;
            o = xa[h*256 + i] * cs[ii] + xa[h*256 + ii] * cs[128 + ii];
        }
        kc[(size_t)slot * DD + c] = o;
    }
}

extern "C" void kernel_launch(void* const* d_in, const int* in_sizes, int n_in,
                              void* d_out, int out_size, void* d_ws, size_t ws_size,
                              hipStream_t stream) {
    (void)in_sizes; (void)n_in; (void)out_size; (void)d_ws; (void)ws_size;

    const float* hidden  = (const float*)d_in[0];
    const float* weight  = (const float*)d_in[1];
    const float* wscale  = (const float*)d_in[2];
    const float* ape     = (const float*)d_in[3];
    const float* norm_w  = (const float*)d_in[4];
    const float* cs      = (const float*)d_in[5];
    // d_in[6] state_cache, d_in[7] kv_cache: initial contents fully overwritten
    const int* positions = (const int*)d_in[8];
    const int* out_loc   = (const int*)d_in[9];
    const int* t2r       = (const int*)d_in[10];
    const int* btab      = (const int*)d_in[11];
    const int* kvmap     = (const int*)d_in[12];

    float* out    = (float*)d_out;
    float* d_kv    = out;                               // 8192*512
    float* d_score = out + (size_t)TT * DD;             // 8192*512
    float* d_sc    = out + (size_t)TT * TWOD;           // 8192*1024
    float* d_kc    = out + (size_t)2 * TT * TWOD;       // 2048*512

    dim3 grid(TT / WG_M, TWOD / WG_N);                  // (64, 8)
    gemm_fused_kernel<<<grid, 256, 0, stream>>>(hidden, weight, wscale, ape,
                                                positions, out_loc,
                                                d_kv, d_score, d_sc);
    compose_kernel<<<TT, 256, 0, stream>>>(d_sc, norm_w, cs, positions, t2r,
                                           btab, kvmap, d_kc);
}